// TFRTokenizer_24129126269380
// MI455X (gfx1250) — compile-verified
//
#include <hip/hip_runtime.h>

typedef __attribute__((ext_vector_type(2))) float v2f;
typedef __attribute__((ext_vector_type(8))) float v8f;

#define C_DIM      32
#define F_DIM      128
#define T_DIM      20000
#define NF         32          // F / PATCH_FREQ
#define P_CF       2500        // patches per (c, fb) == T / PATCH_TIME
#define EMBED      32
#define NTOK       64
#define FULL_TILES 156         // full 16-patch tiles; tile 156 holds the last 4 patches
#define NWAVES     8
#define LDSTR      36          // padded float stride (16B aligned, bank-conflict free)

template<bool TAIL>
__device__ __forceinline__ void do_tile(
    const int tile,
    const float* __restrict__ tfr, const size_t rowBase,
    float* myA, float* myZ, const float* cbl,
    const v2f (&bW)[2][8], const float (&bb)[2], const float (&esq)[4],
    const int lane, const int l15, const int hi,
    int* __restrict__ tok_out, float* __restrict__ z_out, const size_t cfBase)
{
    const int t0 = tile * 128;

    // ---- stage 16 patches: 4 tfr rows x 128 contiguous floats, b128 coalesced.
    // Guard-free in the fast path; tail tile has only 32 valid floats per row.
    #pragma unroll
    for (int pf = 0; pf < 4; ++pf) {
        float4 v = make_float4(0.f, 0.f, 0.f, 0.f);
        if (!TAIL || lane < 8)
            v = *(const float4*)(tfr + rowBase + (size_t)pf * T_DIM + t0 + lane * 4);
        // patch p = lane>>1, within-patch s = pf*8 + (lane&1)*4
        *(float4*)(myA + (lane >> 1) * LDSTR + pf * 8 + (lane & 1) * 4) = v;
    }
    asm volatile("s_wait_dscnt 0x0" ::: "memory");

    // ---- GEMM1: z(16x32) = patches(16x32) . W^T   (8 K-steps of K=4, 2 N-tiles)
    v8f az0 = {0.f,0.f,0.f,0.f,0.f,0.f,0.f,0.f};
    v8f az1 = az0;
    #pragma unroll
    for (int kk = 0; kk < 8; ++kk) {
        const int k = kk * 4 + hi * 2;
        v2f a = *(const v2f*)(myA + l15 * LDSTR + k);
        az0 = __builtin_amdgcn_wmma_f32_16x16x4_f32(false, a, false, bW[0][kk],
                                                    (short)0, az0, false, false);
        az1 = __builtin_amdgcn_wmma_f32_16x16x4_f32(false, a, false, bW[1][kk],
                                                    (short)0, az1, false, false);
    }

    // ---- bias add + z tile to wave-private LDS (single base -> ds_store_2addr)
    #pragma unroll
    for (int r = 0; r < 8; ++r) {
        float* zp = myZ + (r + hi * 8) * LDSTR + l15;
        zp[0]  = az0[r] + bb[0];
        zp[16] = az1[r] + bb[1];
    }
    asm volatile("s_wait_dscnt 0x0" ::: "memory");

    // ---- z -> global: 512 contiguous floats per tile, b128 fully coalesced
    {
        const size_t zb = (cfBase + (size_t)tile * 16) * EMBED;
        #pragma unroll
        for (int j = 0; j < (TAIL ? 1 : 4); ++j) {   // tail: only 4 patches (j==0)
            const int i = lane + 32 * j;             // patch = i>>3, e4 = i&7
            float4 v = *(const float4*)(myZ + (i >> 3) * LDSTR + (i & 7) * 4);
            *(float4*)(z_out + zb + (size_t)i * 4) = v;
        }
    }

    // ---- GEMM2: cross(16x64) = z(16x32) . codebook^T   (4 N-tiles)
    v8f ac[4];
    #pragma unroll
    for (int nt = 0; nt < 4; ++nt) ac[nt] = (v8f){0.f,0.f,0.f,0.f,0.f,0.f,0.f,0.f};
    #pragma unroll
    for (int kk = 0; kk < 8; ++kk) {
        const int k = kk * 4 + hi * 2;
        v2f a = *(const v2f*)(myZ + l15 * LDSTR + k);
        #pragma unroll
        for (int nt = 0; nt < 4; ++nt) {
            v2f bf = *(const v2f*)(cbl + (nt * 16 + l15) * LDSTR + k);
            ac[nt] = __builtin_amdgcn_wmma_f32_16x16x4_f32(false, a, false, bf,
                                                           (short)0, ac[nt], false, false);
        }
    }

    // ---- argmax_k (2*cross - e_sq): 4 local tokens, then 16-lane shfl reduce
    #pragma unroll
    for (int r = 0; r < (TAIL ? 4 : 8); ++r) {
        float best = -3.402823466e+38f;
        int   btok = 0;
        #pragma unroll
        for (int nt = 0; nt < 4; ++nt) {
            const float sc  = 2.f * ac[nt][r] - esq[nt];
            const int   tkn = nt * 16 + l15;
            if (sc > best || (sc == best && tkn < btok)) { best = sc; btok = tkn; }
        }
        #pragma unroll
        for (int off = 8; off >= 1; off >>= 1) {     // stays within each 16-lane half
            const float ov = __shfl_xor(best, off, 32);
            const int   ot = __shfl_xor(btok, off, 32);
            if (ov > best || (ov == best && ot < btok)) { best = ov; btok = ot; }
        }
        if (l15 == 0 && (!TAIL || hi == 0))
            tok_out[cfBase + tile * 16 + r + hi * 8] = btok;
    }
}

__global__ __launch_bounds__(256) void tfr_tokenizer_wmma(
    const float* __restrict__ tfr,      // (32,128,20000)
    const float* __restrict__ W,        // (32,32)
    const float* __restrict__ bias,     // (32,)
    const float* __restrict__ codebook, // (64,32)
    int*   __restrict__ tok_out,        // (32,80000)
    float* __restrict__ z_out)          // (32,80000,32)
{
    __shared__ float cbl[NTOK * LDSTR];            // padded codebook
    __shared__ float tA[NWAVES][16 * LDSTR];       // per-wave patch tile (A for GEMM1)
    __shared__ float tZ[NWAVES][16 * LDSTR];       // per-wave z tile   (A for GEMM2)

    const int tid  = threadIdx.x;
    const int wave = tid >> 5;
    const int lane = tid & 31;
    const int l15  = lane & 15;
    const int hi   = lane >> 4;

    const int c  = blockIdx.x >> 5;
    const int fb = blockIdx.x & 31;

    // ---- stage codebook once per block (b128 granularity, padded stride)
    #pragma unroll
    for (int i = tid; i < (NTOK * EMBED) / 4; i += 256) {
        float4 v = ((const float4*)codebook)[i];
        *(float4*)(cbl + (i >> 3) * LDSTR + (i & 7) * 4) = v;  // token = i>>3, e4 = i&7
    }
    __syncthreads();

    // ---- per-lane e_sq for its 4 tokens (token = nt*16 + l15), b128 reads
    float esq[4];
    #pragma unroll
    for (int nt = 0; nt < 4; ++nt) {
        const float4* row = (const float4*)(cbl + (nt * 16 + l15) * LDSTR);
        float s = 0.f;
        #pragma unroll
        for (int q = 0; q < 8; ++q) {
            float4 v = row[q];
            s += v.x * v.x + v.y * v.y + v.z * v.z + v.w * v.w;
        }
        esq[nt] = s;
    }

    // ---- loop-invariant W B-fragments + bias (registers)
    v2f bW[2][8];
    float bb[2];
    #pragma unroll
    for (int nt = 0; nt < 2; ++nt) {
        const int e = nt * 16 + l15;
        bb[nt] = bias[e];
        #pragma unroll
        for (int kk = 0; kk < 8; ++kk)
            bW[nt][kk] = *(const v2f*)(W + e * EMBED + kk * 4 + hi * 2);
    }

    const size_t rowBase = ((size_t)c * F_DIM + fb * 4) * T_DIM;
    const size_t cfBase  = (size_t)c * (NF * P_CF) + (size_t)fb * P_CF;

    // guard-free main loop over full tiles; tail tile handled by its owner wave
    for (int tile = wave; tile < FULL_TILES; tile += NWAVES)
        do_tile<false>(tile, tfr, rowBase, tA[wave], tZ[wave], cbl,
                       bW, bb, esq, lane, l15, hi, tok_out, z_out, cfBase);
    if (wave == (FULL_TILES % NWAVES))
        do_tile<true>(FULL_TILES, tfr, rowBase, tA[wave], tZ[wave], cbl,
                      bW, bb, esq, lane, l15, hi, tok_out, z_out, cfBase);
}

extern "C" void kernel_launch(void* const* d_in, const int* in_sizes, int n_in,
                              void* d_out, int out_size, void* d_ws, size_t ws_size,
                              hipStream_t stream) {
    const float* tfr = (const float*)d_in[0];   // (32,128,20000) f32
    const float* W   = (const float*)d_in[1];   // (32,32) f32
    const float* b   = (const float*)d_in[2];   // (32,) f32
    const float* cb  = (const float*)d_in[3];   // (64,32) f32
    int*   tok = (int*)d_out;                               // 2,560,000 int32 token ids
    float* z   = (float*)d_out + (size_t)C_DIM * NF * P_CF; // then 81,920,000 f32 z
    tfr_tokenizer_wmma<<<dim3(C_DIM * NF), dim3(256), 0, stream>>>(tfr, W, b, cb, tok, z);
}